// compute_G_75505525063760
// MI455X (gfx1250) — compile-verified
//
#include <hip/hip_runtime.h>

// G = DV2_H @ diag(W) @ invDE_HT_DV2
//   A = DV2_H            [N x E] row-major, f32   (N=16384, E=128)
//   B = invDE_HT_DV2     [E x N] row-major, f32
//   W = per-hyperedge    [E] f32
//   C = G                [N x N] f32
//
// Store-bound GEMM (1 GiB output). f32 WMMA path: V_WMMA_F32_16X16X4_F32,
// K=128 -> 32 chained WMMAs per 16x16 tile. Wave tile 32x32 (2x2 WMMA tiles),
// block tile 64x128 (8 waves, 2x4).

typedef __attribute__((ext_vector_type(2))) float v2f;
typedef __attribute__((ext_vector_type(8))) float v8f;

#define NN 16384
#define EE 128

__global__ __launch_bounds__(256) void hgnn_g_wmma_f32(
    const float* __restrict__ A,   // [N,E]
    const float* __restrict__ B,   // [E,N]
    const float* __restrict__ W,   // [E]
    float* __restrict__ C)         // [N,N]
{
    const int lane = threadIdx.x & 31;
    const int wave = threadIdx.x >> 5;
    const int half = lane >> 4;    // 0: lanes 0-15, 1: lanes 16-31
    const int l16  = lane & 15;

    const int waveM = wave >> 2;   // 0..1
    const int waveN = wave & 3;    // 0..3

    const int tileM0 = blockIdx.y * 64  + waveM * 32;  // row base of wave tile
    const int tileN0 = blockIdx.x * 128 + waveN * 32;  // col base of wave tile

    v8f acc00 = {}, acc01 = {}, acc10 = {}, acc11 = {};

    // Per-lane base pointers.
    // A-fragment layout (32-bit A 16x4): lane l16 -> row M=l16,
    // lane-half selects K pair: half 0 -> K={k,k+1}, half 1 -> K={k+2,k+3}.
    const float* Arow0 = A + (long)(tileM0 +      l16) * EE;
    const float* Arow1 = A + (long)(tileM0 + 16 + l16) * EE;
    // B-fragment layout (4x16): lane l16 -> col N=l16, same K-pair split.
    const float* Bcol0 = B + (tileN0 +      l16);
    const float* Bcol1 = B + (tileN0 + 16 + l16);

#pragma unroll 4
    for (int k = 0; k < EE; k += 4) {
        const int kb = k + 2 * half;   // this lane's K base (even -> 8B aligned)

        // A fragments: contiguous float2 per lane (global_load_b64)
        v2f a0 = *(const v2f*)(Arow0 + kb);
        v2f a1 = *(const v2f*)(Arow1 + kb);

        // diag(W): per-lane pair of weights for K=kb, kb+1 (L2/L0-resident)
        v2f w = *(const v2f*)(W + kb);

        // B fragments: two stride-N loads per subtile, scaled by W in-register
        v2f b0, b1;
        b0.x = Bcol0[(long)kb       * NN] * w.x;
        b0.y = Bcol0[(long)(kb + 1) * NN] * w.y;
        b1.x = Bcol1[(long)kb       * NN] * w.x;
        b1.y = Bcol1[(long)(kb + 1) * NN] * w.y;

        // D = A x B + C  (f32, 16x16x4)
        acc00 = __builtin_amdgcn_wmma_f32_16x16x4_f32(false, a0, false, b0,
                                                      (short)0, acc00, false, false);
        acc01 = __builtin_amdgcn_wmma_f32_16x16x4_f32(false, a0, false, b1,
                                                      (short)0, acc01, false, false);
        acc10 = __builtin_amdgcn_wmma_f32_16x16x4_f32(false, a1, false, b0,
                                                      (short)0, acc10, false, false);
        acc11 = __builtin_amdgcn_wmma_f32_16x16x4_f32(false, a1, false, b1,
                                                      (short)0, acc11, false, false);
    }

    // C/D 16x16 f32 layout: VGPR j -> row (base + j + 8*half), col (base + l16)
    {
        v8f* accs[2][2] = { { &acc00, &acc01 }, { &acc10, &acc11 } };
#pragma unroll
        for (int mi = 0; mi < 2; ++mi) {
#pragma unroll
            for (int ni = 0; ni < 2; ++ni) {
                const long rowBase = tileM0 + mi * 16 + half * 8;
                const long colBase = tileN0 + ni * 16 + l16;
                float* Cp = C + rowBase * (long)NN + colBase;
                v8f av = *accs[mi][ni];
#pragma unroll
                for (int j = 0; j < 8; ++j)
                    Cp[(long)j * NN] = av[j];
            }
        }
    }
}

extern "C" void kernel_launch(void* const* d_in, const int* in_sizes, int n_in,
                              void* d_out, int out_size, void* d_ws, size_t ws_size,
                              hipStream_t stream)
{
    (void)in_sizes; (void)n_in; (void)out_size; (void)d_ws; (void)ws_size;

    const float* A = (const float*)d_in[0];  // DV2_H        [N,E]
    const float* B = (const float*)d_in[1];  // invDE_HT_DV2 [E,N]
    const float* W = (const float*)d_in[2];  // W            [E]
    float*       C = (float*)d_out;          // G            [N,N]

    dim3 block(256, 1, 1);
    dim3 grid(NN / 128, NN / 64, 1);  // x: col tiles (128 wide), y: row tiles (64 tall)
    hipLaunchKernelGGL(hgnn_g_wmma_f32, grid, block, 0, stream, A, B, W, C);
}